// BERTCRF_43001212567562
// MI455X (gfx1250) — compile-verified
//
#include <hip/hip_runtime.h>
#include <hip/hip_bf16.h>
#include <math.h>

// Problem sizes (fixed by the reference)
#define B_  64
#define T_  512
#define H_  768
#define L_  9
#define LP  16   // label dim padded to WMMA N=16

typedef float v2f __attribute__((ext_vector_type(2)));
typedef float v8f __attribute__((ext_vector_type(8)));

// ---------------------------------------------------------------------------
// Kernel 1: emissions[b,t,l] = hidden[b,t,:] @ W[:,l] + bias[l]
// Stored padded to LP=16 columns in workspace.
// One wave per 16-row tile of the flattened (B*T, H) matrix, K-loop of
// V_WMMA_F32_16X16X4_F32.
//   A operand: loaded directly in WMMA layout (lane l: K=kb,kb+1; lane l+16:
//     K=kb+2,kb+3) -> one float2 per lane, 16 contiguous B per row, no selects.
//   B operand: W staged in LDS pre-swizzled into B-operand *pair* layout so
//     each lane's (v0,v1) is one aligned ds_load_b64 with immediate offsets.
//   Bias folded into the initial WMMA accumulator (D = A*B + C).
// ---------------------------------------------------------------------------
__global__ __launch_bounds__(256)
void bertcrf_emis_wmma(const float* __restrict__ hidden,
                       const float* __restrict__ W,
                       const float* __restrict__ bias,
                       float* __restrict__ em) {
  // Pair layout: Wlds[((kb>>2)*2 + h)*32 + n*2 + p] = Wpad[kb + 2h + p][n]
  __shared__ float Wlds[H_ * LP];   // 48 KB
  __shared__ float biasL[LP];
  const int tid = threadIdx.x;

  for (int idx = tid; idx < H_ * LP; idx += 256) {
    const int pairblk = idx >> 5;      // (kb>>2)*2 + h
    const int within  = idx & 31;
    const int n       = within >> 1;
    const int p       = within & 1;
    const int k       = (pairblk >> 1) * 4 + (pairblk & 1) * 2 + p;
    Wlds[idx] = (n < L_) ? W[k * L_ + n] : 0.0f;
  }
  if (tid < LP) biasL[tid] = (tid < L_) ? bias[tid] : 0.0f;
  __syncthreads();

  const int  lane = tid & 31;
  const int  wave = tid >> 5;
  const bool hi   = (lane & 16) != 0;
  const int  col  = lane & 15;

  const int tile = blockIdx.x * 8 + wave;  // 256 blocks * 8 waves = 2048 tiles
  const int row0 = tile * 16;              // covers B_*T_ = 32768 rows exactly

  // Per-lane A pointer, already shifted by the half-wave K offset.
  const float* aptr = hidden + (size_t)(row0 + col) * H_ + (hi ? 2 : 0);

  // Per-lane B base inside the pair-swizzled LDS image.
  const float* bptr = &Wlds[(hi ? 32 : 0) + col * 2];

  // Fold the bias in as the initial C matrix (same value down each column).
  const float bb = biasL[col];
  v8f acc;
  #pragma unroll
  for (int v = 0; v < 8; ++v) acc[v] = bb;

  #pragma unroll 8
  for (int kb = 0; kb < H_; kb += 4) {
    const v2f a  = *reinterpret_cast<const v2f*>(aptr + kb);          // global b64
    const v2f bv = *reinterpret_cast<const v2f*>(bptr + (kb >> 2) * 64); // ds b64
    // (neg_a, A, neg_b, B, c_mod, C, reuse_a, reuse_b)
    acc = __builtin_amdgcn_wmma_f32_16x16x4_f32(false, a, false, bv,
                                                (short)0, acc, false, false);
  }

  // D layout: VGPR v -> M=v (lanes 0-15) / M=8+v (lanes 16-31), N=lane%16
  const int mbase = row0 + (hi ? 8 : 0);
  #pragma unroll
  for (int v = 0; v < 8; ++v) {
    em[(size_t)(mbase + v) * LP + col] = acc[v];
  }
}

// ---------------------------------------------------------------------------
// Kernel 2: per-batch CRF negative log-likelihood.
// One wave per batch. Emissions staged to LDS with the CDNA5 async
// global->LDS path (GLOBAL_LOAD_ASYNC_TO_LDS_B128, ASYNCcnt-tracked, no VGPR
// round trip), then a 511-step serial logsumexp recursion entirely out of
// LDS/registers. Lanes 0..8 own the 9 tag states.
// ---------------------------------------------------------------------------
__global__ __launch_bounds__(32)
void bertcrf_crf(const float* __restrict__ em,
                 const float* __restrict__ startT,
                 const float* __restrict__ endT,
                 const float* __restrict__ trans,
                 const int* __restrict__ mask,
                 const int* __restrict__ labels,
                 float* __restrict__ out) {
  __shared__ float emL[T_ * LP];  // 32 KB: this batch's padded emissions
  __shared__ int   mkL[T_];

  const int b    = blockIdx.x;
  const int lane = threadIdx.x;

  const float* emg = em + (size_t)b * T_ * LP;
  const int*   mkg = mask + b * T_;
  const int*   lbg = labels + b * T_;

  // Async copy 32 KB emissions: 2048 float4 / 32 lanes = 64 b128 transfers.
  for (int idx = lane; idx < (T_ * LP) / 4; idx += 32) {
    const unsigned int       ldsa = (unsigned int)(uintptr_t)(&emL[idx * 4]);
    const unsigned long long ga   = (unsigned long long)(uintptr_t)(emg + idx * 4);
    asm volatile("global_load_async_to_lds_b128 %0, %1, off"
                 :: "v"(ldsa), "v"(ga) : "memory");
  }
  for (int t = lane; t < T_; t += 32) mkL[t] = mkg[t];
  asm volatile("s_wait_asynccnt 0" ::: "memory");
  __syncthreads();

  const int j = (lane < L_) ? lane : 0;
  float tc[L_];  // transitions column j: trans[i][j]
  #pragma unroll
  for (int i = 0; i < L_; ++i) tc[i] = trans[i * L_ + j];

  // ---- gold score, parallel over t then wave-reduced ----
  float sp   = 0.0f;
  int   msum = 0;
  for (int t = lane; t < T_; t += 32) {
    const int m = mkL[t];
    msum += m;
    if (t >= 1 && m > 0) {
      int tp = lbg[t - 1]; tp = tp < 0 ? 0 : tp;
      int tg = lbg[t];     tg = tg < 0 ? 0 : tg;
      sp += trans[tp * L_ + tg] + emL[t * LP + tg];
    }
  }
  #pragma unroll
  for (int off = 16; off > 0; off >>= 1) {
    sp   += __shfl_down(sp, off, 32);
    msum += __shfl_down(msum, off, 32);
  }
  sp   = __shfl(sp, 0, 32);
  msum = __shfl(msum, 0, 32);

  float score;
  {
    int t0 = lbg[0]; t0 = t0 < 0 ? 0 : t0;
    int se = msum - 1; if (se < 0) se = 0;
    int lt = lbg[se]; lt = lt < 0 ? 0 : lt;
    score = startT[t0] + emL[t0] + sp + endT[lt];
  }

  // ---- forward recursion (serial over T, lanes = states) ----
  const float NEG = -1.0e30f;
  float alpha = (lane < L_) ? (startT[j] + emL[j]) : NEG;
  for (int t = 1; t < T_; ++t) {
    const float emj = emL[t * LP + j];
    const int   m   = mkL[t];
    float vals[L_];
    float vmax = NEG;
    #pragma unroll
    for (int i = 0; i < L_; ++i) {
      const float v = __shfl(alpha, i, 32) + tc[i];
      vals[i] = v;
      vmax = fmaxf(vmax, v);
    }
    float s = 0.0f;
    #pragma unroll
    for (int i = 0; i < L_; ++i) s += __expf(vals[i] - vmax);
    const float na = vmax + __logf(s) + emj;
    if (lane < L_ && m > 0) alpha = na;
  }

  // ---- log_z = logsumexp(alpha + end_transitions) ----
  const float v  = (lane < L_) ? (alpha + endT[j]) : NEG;
  float vm = v;
  #pragma unroll
  for (int off = 16; off > 0; off >>= 1) vm = fmaxf(vm, __shfl_down(vm, off, 32));
  vm = __shfl(vm, 0, 32);
  float s = __expf(v - vm);  // inactive lanes underflow to 0
  #pragma unroll
  for (int off = 16; off > 0; off >>= 1) s += __shfl_down(s, off, 32);

  if (lane == 0) {
    const float logz = vm + __logf(s);
    atomicAdd(out, (logz - score) * (1.0f / (float)B_));  // loss = mean(logz - score)
  }
}

// ---------------------------------------------------------------------------
extern "C" void kernel_launch(void* const* d_in, const int* in_sizes, int n_in,
                              void* d_out, int out_size, void* d_ws, size_t ws_size,
                              hipStream_t stream) {
  (void)in_sizes; (void)n_in; (void)out_size; (void)ws_size;

  const float* hidden = (const float*)d_in[0];
  const float* W      = (const float*)d_in[1];
  const float* bias   = (const float*)d_in[2];
  const float* startT = (const float*)d_in[3];
  const float* endT   = (const float*)d_in[4];
  const float* trans  = (const float*)d_in[5];
  const int*   mask   = (const int*)d_in[6];
  const int*   labels = (const int*)d_in[7];

  float* em = (float*)d_ws;  // B_*T_*LP fp32 = 2 MB padded emissions

  hipMemsetAsync(d_out, 0, sizeof(float), stream);

  bertcrf_emis_wmma<<<dim3(256), dim3(256), 0, stream>>>(hidden, W, bias, em);
  bertcrf_crf<<<dim3(B_), dim3(32), 0, stream>>>(em, startT, endT, trans,
                                                 mask, labels, (float*)d_out);
}